// TopKSegments_62715112456633
// MI455X (gfx1250) — compile-verified
//
#include <hip/hip_runtime.h>
#include <stdint.h>

#define SEQ_LEN      8388608
#define NUM_SEGMENTS 32
#define MIN_DIST     256
#define CHUNK        2048
#define NCHUNKS      (SEQ_LEN / CHUNK)     /* 4096 */
#define P1_BLOCK     256
#define CHUNKS_PER_BLOCK 4
#define P1_GRID      (NCHUNKS / CHUNKS_PER_BLOCK)  /* 1024 */

#define NEGINF (-__builtin_inff())

/* ---------------- CDNA5 async global->LDS path (guarded) ---------------- */
#if defined(__has_builtin)
#if __has_builtin(__builtin_amdgcn_global_load_async_to_lds_b128)
#define ASYNC_LDS 1
#endif
#endif

#ifdef ASYNC_LDS
/* Signature (from clang diagnostic): (v4i AS1*, v4i AS3*, imm i32, imm i32) */
typedef int v4i __attribute__((vector_size(4 * sizeof(int))));
typedef __attribute__((address_space(1))) v4i* as1_v4i_ptr;
typedef __attribute__((address_space(3))) v4i* as3_v4i_ptr;
#endif

__device__ __forceinline__ void async_copy16(const float* g, float* l) {
#ifdef ASYNC_LDS
  /* global_load_async_to_lds_b128: per-lane 16B, tracked by ASYNCcnt */
  __builtin_amdgcn_global_load_async_to_lds_b128(
      (as1_v4i_ptr)(void*)g, (as3_v4i_ptr)(void*)l, 0, 0);
#else
  *(float4*)l = *(const float4*)g;
#endif
}

#ifdef ASYNC_LDS
#if __has_builtin(__builtin_amdgcn_s_wait_asynccnt)
#define WAIT_ASYNC(n) __builtin_amdgcn_s_wait_asynccnt(n)
#else
#define WAIT_ASYNC(n) asm volatile("s_wait_asynccnt " #n ::: "memory")
#endif
#else
#define WAIT_ASYNC(n) do {} while (0)
#endif

/* ---------------- wave32 argmax (lowest-index tie-break) ---------------- */
__device__ __forceinline__ void wave_argmax(float& v, int& i) {
  #pragma unroll
  for (int off = 16; off > 0; off >>= 1) {
    float v2 = __shfl_down(v, off, 32);
    int   i2 = __shfl_down(i, off, 32);
    if (v2 > v || (v2 == v && i2 < i)) { v = v2; i = i2; }
  }
}

/* =======================================================================
 * Pass 1: per-chunk (max, argmax) of masked scores.
 * Double-buffered async global->LDS streaming; one full read of inputs.
 * ======================================================================= */
__global__ void __launch_bounds__(P1_BLOCK)
chunkmax_kernel(const float* __restrict__ scores,
                const unsigned char* __restrict__ mask,
                float* __restrict__ cv, int* __restrict__ ci) {
  __shared__ float sbuf[2][CHUNK];   /* 16 KB double buffer */
  __shared__ float wv[P1_BLOCK / 32];
  __shared__ int   wi[P1_BLOCK / 32];

  const int t    = threadIdx.x;
  const int lane = t & 31;
  const int wave = t >> 5;
  const int c0   = blockIdx.x * CHUNKS_PER_BLOCK;

  /* prologue: stage chunk c0 into buffer 0 */
  {
    const float* g = scores + (size_t)c0 * CHUNK + t * 8;
    float* l = &sbuf[0][t * 8];
    async_copy16(g, l);
    async_copy16(g + 4, l + 4);
  }

  for (int k = 0; k < CHUNKS_PER_BLOCK; ++k) {
    const int c    = c0 + k;
    const int base = c * CHUNK;

    if (k + 1 < CHUNKS_PER_BLOCK) {
      /* prefetch next chunk into the other buffer, then wait for current */
      const float* g = scores + (size_t)base + CHUNK + t * 8;
      float* l = &sbuf[(k + 1) & 1][t * 8];
      async_copy16(g, l);
      async_copy16(g + 4, l + 4);
      WAIT_ASYNC(2);   /* 2 newest (next chunk) may remain outstanding */
    } else {
      WAIT_ASYNC(0);
    }
    __syncthreads();

    /* 8 contiguous elements per thread -> lowest-index tie-break holds */
    const unsigned long long m8 =
        *(const unsigned long long*)(mask + (size_t)base + t * 8);
    const float* lb = &sbuf[k & 1][t * 8];

    float bv = ((m8 & 0xffull) != 0) ? lb[0] : NEGINF;
    int   bi = base + t * 8;
    #pragma unroll
    for (int j = 1; j < 8; ++j) {
      float v = (((m8 >> (8 * j)) & 0xffull) != 0) ? lb[j] : NEGINF;
      if (v > bv) { bv = v; bi = base + t * 8 + j; }
    }

    wave_argmax(bv, bi);
    if (lane == 0) { wv[wave] = bv; wi[wave] = bi; }
    __syncthreads();
    if (t == 0) {
      float v = wv[0]; int ix = wi[0];
      #pragma unroll
      for (int w = 1; w < P1_BLOCK / 32; ++w)
        if (wv[w] > v || (wv[w] == v && wi[w] < ix)) { v = wv[w]; ix = wi[w]; }
      cv[c] = v; ci[c] = ix;
    }
    __syncthreads();   /* protect sbuf / wv reuse */
  }
}

/* =======================================================================
 * Pass 2: 32 serial NMS rounds, chunk table resident in LDS.
 * Each round: argmax over 4096 (val,idx) entries, then recompute the <=2
 * chunks touched by the +/-256 window against the selected-window list.
 * ======================================================================= */
__device__ __forceinline__ float masked_score(int p,
                                              const float* __restrict__ s,
                                              const unsigned char* __restrict__ m,
                                              const int* sup, int nsel) {
  float v = m[p] ? s[p] : NEGINF;
  for (int j = 0; j < nsel; ++j) {
    int d = p - sup[j];
    if (d < 0) d = -d;
    if (d <= MIN_DIST) v = NEGINF;
  }
  return v;
}

__global__ void __launch_bounds__(1024)
nms_kernel(const float* __restrict__ scores,
           const unsigned char* __restrict__ mask,
           const float* __restrict__ cv, const int* __restrict__ ci,
           long long* __restrict__ out) {
  __shared__ float vals[NCHUNKS];    /* 16 KB */
  __shared__ int   idxs[NCHUNKS];    /* 16 KB */
  __shared__ float wv[32];
  __shared__ int   wi[32];
  __shared__ int   sup[NUM_SEGMENTS];
  __shared__ int   besti_s;

  const int t    = threadIdx.x;
  const int lane = t & 31;
  const int wave = t >> 5;

  for (int i = t; i < NCHUNKS; i += 1024) { vals[i] = cv[i]; idxs[i] = ci[i]; }
  __syncthreads();

  for (int it = 0; it < NUM_SEGMENTS; ++it) {
    /* ---- block-wide argmax over the chunk table ---- */
    float bv = vals[t];
    int   bi = idxs[t];
    #pragma unroll
    for (int i = 1; i < NCHUNKS / 1024; ++i) {
      float v  = vals[t + i * 1024];
      int   ix = idxs[t + i * 1024];
      if (v > bv || (v == bv && ix < bi)) { bv = v; bi = ix; }
    }
    wave_argmax(bv, bi);
    if (lane == 0) { wv[wave] = bv; wi[wave] = bi; }
    __syncthreads();
    if (wave == 0) {
      float v = wv[lane]; int ix = wi[lane];
      wave_argmax(v, ix);
      if (lane == 0) {
        besti_s = ix;
        sup[it] = ix;                                    /* window center */
        out[it] = (v > NEGINF) ? (long long)ix : -1ll;   /* -1 if exhausted */
      }
    }
    __syncthreads();

    /* ---- suppress: recompute the <=2 chunks under the +/-256 window ---- */
    const int mi = besti_s;
    int lo = mi - MIN_DIST; if (lo < 0) lo = 0;
    int hi = mi + MIN_DIST; if (hi > SEQ_LEN - 1) hi = SEQ_LEN - 1;
    const int clo = lo / CHUNK, chi = hi / CHUNK;
    for (int c = clo; c <= chi; ++c) {
      const int base = c * CHUNK;
      const int p0   = base + t * 2;     /* 2 contiguous elems per thread */
      float v0 = masked_score(p0,     scores, mask, sup, it + 1);
      float v1 = masked_score(p0 + 1, scores, mask, sup, it + 1);
      float cbv = v0; int cbi = p0;
      if (v1 > cbv) { cbv = v1; cbi = p0 + 1; }
      wave_argmax(cbv, cbi);
      if (lane == 0) { wv[wave] = cbv; wi[wave] = cbi; }
      __syncthreads();
      if (t == 0) {
        float v = wv[0]; int ix = wi[0];
        #pragma unroll
        for (int w = 1; w < 32; ++w)
          if (wv[w] > v || (wv[w] == v && wi[w] < ix)) { v = wv[w]; ix = wi[w]; }
        vals[c] = v; idxs[c] = ix;
      }
      __syncthreads();
    }
  }
}

/* ======================================================================= */
extern "C" void kernel_launch(void* const* d_in, const int* in_sizes, int n_in,
                              void* d_out, int out_size, void* d_ws, size_t ws_size,
                              hipStream_t stream) {
  (void)in_sizes; (void)n_in; (void)out_size; (void)ws_size;
  const float*         scores = (const float*)d_in[0];
  const unsigned char* mask   = (const unsigned char*)d_in[1];

  float* cv = (float*)d_ws;                                   /* 16 KB */
  int*   ci = (int*)((char*)d_ws + NCHUNKS * sizeof(float));  /* 16 KB */
  long long* out = (long long*)d_out;                         /* int64 indices */

  chunkmax_kernel<<<P1_GRID, P1_BLOCK, 0, stream>>>(scores, mask, cv, ci);
  nms_kernel<<<1, 1024, 0, stream>>>(scores, mask, cv, ci, out);
}